// SAGE_9371618640569
// MI455X (gfx1250) — compile-verified
//
#include <hip/hip_runtime.h>

// ---------------------------------------------------------------------------
// GraphSAGE (3x SAGEConv + BN + ReLU) for MI455X / gfx1250, fp32 throughout.
//   layer: agg[dst] += h[src]           (global_atomic_add_f32, L2-resident)
//          out = (agg*inv) @ Wl + h @ Wr (+bias)   via V_WMMA_F32_16X16X4_F32
//          BN(train-mode, biased var) + ReLU for layers 0,1
// Weights are pre-packed into WMMA B-fragment order so the GEMM inner loop is
// 1x global_load_b128 + 4x ds_load_b64 + 4x v_wmma per K-step, unconditional.
// ---------------------------------------------------------------------------

#define IN_D   128
#define HID_D  128
#define OUT_D  40
#define BN_EPS 1e-5f
#define LDS_STRIDE (HID_D + 4)   // 132: 16 A-rows -> distinct LDS banks
#define KSTEPS (HID_D / 4)       // 32

typedef __attribute__((ext_vector_type(2))) float v2f;
typedef __attribute__((ext_vector_type(8))) float v8f;

__device__ __forceinline__ void atomAddF(float* p, float v) {
    unsafeAtomicAdd(p, v);   // global_atomic_add_f32 (no CAS loop)
}

// ---------------------------------------------------------------------------
__global__ void zero_f32(float* __restrict__ p, size_t n) {
    size_t i = (size_t)blockIdx.x * blockDim.x + threadIdx.x;
    if (i < n) p[i] = 0.0f;
}

__global__ void count_deg(const long long* __restrict__ ei, float* __restrict__ cnt, int E) {
    int e = blockIdx.x * blockDim.x + threadIdx.x;
    if (e < E) atomAddF(&cnt[(int)ei[(size_t)E + e]], 1.0f);
}

__global__ void make_inv(float* __restrict__ cnt, int n) {
    int i = blockIdx.x * blockDim.x + threadIdx.x;
    if (i < n) cnt[i] = 1.0f / fmaxf(cnt[i], 1.0f);
}

// agg[dst][c..c+3] += h[src][c..c+3]; one thread per (edge, 4-channel chunk)
__global__ void scatter_add(const float* __restrict__ h,
                            const long long* __restrict__ ei,
                            float* __restrict__ agg, int E) {
    const int chunks = HID_D / 4;  // 32
    size_t idx = (size_t)blockIdx.x * blockDim.x + threadIdx.x;
    if (idx >= (size_t)E * chunks) return;
    int e = (int)(idx / chunks);
    int q = (int)(idx % chunks) * 4;
    int src = (int)ei[e];
    int dst = (int)ei[(size_t)E + e];
    const float4 v = *(const float4*)(h + (size_t)src * HID_D + q);
    float* p = agg + (size_t)dst * HID_D + q;
    atomAddF(p + 0, v.x);
    atomAddF(p + 1, v.y);
    atomAddF(p + 2, v.z);
    atomAddF(p + 3, v.w);
}

// ---------------------------------------------------------------------------
// Pre-pack (Wl,Wr) into per-lane B-fragment order, zero-padded past ncols:
//   pack[t][s][lane] = { Wl[ak][col], Wl[ak+1][col], Wr[ak][col], Wr[ak+1][col] }
//   col = t*16 + (lane&15), ak = s*4 + 2*(lane>>4)   (CDNA5 32-bit B 4x16 split)
__global__ void pack_w(const float* __restrict__ Wl, const float* __restrict__ Wr,
                       float4* __restrict__ pack, int ntiles, int ncols) {
    int idx = blockIdx.x * blockDim.x + threadIdx.x;
    if (idx >= ntiles * KSTEPS * 32) return;
    int lane = idx & 31;
    int s    = (idx >> 5) & (KSTEPS - 1);
    int t    = idx >> 10;
    int col  = t * 16 + (lane & 15);
    int ak   = s * 4 + 2 * (lane >> 4);
    float4 v = {0.f, 0.f, 0.f, 0.f};
    if (col < ncols) {
        v.x = Wl[(size_t)ak * ncols + col];
        v.y = Wl[(size_t)(ak + 1) * ncols + col];
        v.z = Wr[(size_t)ak * ncols + col];
        v.w = Wr[(size_t)(ak + 1) * ncols + col];
    }
    pack[idx] = v;
}

// ---------------------------------------------------------------------------
// Block = (32, ncol_tiles); grid.x covers 32 rows (2 M-tiles) per block.
// Each wave: two 16x16 output tiles, K=128 in steps of 4.
// Fragment layouts (CDNA5 ISA 7.12.2, 32-bit ops):
//   A: lane(0-15)=row M, hi=lane>>4 selects K pair {ak, ak+1}, ak=k0+2*hi
//   B: same K split, col = n0 + (lane&15)   (from packed array)
//   D: vgpr v -> row v + 8*hi, col = n0 + (lane&15)
__global__ __launch_bounds__(256)
void sage_gemm_wmma(const float* __restrict__ agg,
                    const float* __restrict__ invc,
                    const float* __restrict__ root,
                    const float4* __restrict__ packW,
                    const float* __restrict__ bias,   // nullptr if none
                    float* __restrict__ out,
                    int nrows, int ncols) {
    __shared__ float sMean[32][LDS_STRIDE];
    __shared__ float sRoot[32][LDS_STRIDE];

    const int row0 = blockIdx.x * 32;
    const int lane = threadIdx.x;           // 0..31
    const int wy   = threadIdx.y;           // column-tile id
    const int tid  = wy * 32 + lane;
    const int nthr = blockDim.x * blockDim.y;

    // Cooperative, coalesced staging of 32x128 slabs (mean pre-scaled).
    for (int i = tid; i < 32 * (HID_D / 4); i += nthr) {
        int r = i >> 5;                // / (HID_D/4)
        int c = (i & 31) * 4;
        int g = row0 + r;
        float4 a = {0.f, 0.f, 0.f, 0.f}, x = {0.f, 0.f, 0.f, 0.f};
        float s = 0.f;
        if (g < nrows) {
            s = invc[g];
            a = *(const float4*)(agg  + (size_t)g * HID_D + c);
            x = *(const float4*)(root + (size_t)g * HID_D + c);
        }
        sMean[r][c + 0] = a.x * s; sMean[r][c + 1] = a.y * s;
        sMean[r][c + 2] = a.z * s; sMean[r][c + 3] = a.w * s;
        sRoot[r][c + 0] = x.x;     sRoot[r][c + 1] = x.y;
        sRoot[r][c + 2] = x.z;     sRoot[r][c + 3] = x.w;
    }
    __syncthreads();

    const int nc  = lane & 15;
    const int hi  = lane >> 4;      // 0 or 1
    const int col = wy * 16 + nc;
    const float4* wp = packW + (size_t)wy * KSTEPS * 32 + lane;

    v8f c0 = {}, c1 = {};
    #pragma unroll
    for (int s = 0; s < KSTEPS; ++s) {
        const float4 b = wp[(size_t)s * 32];     // coalesced b128, no predication
        const int ak = s * 4 + 2 * hi;
        v2f bL, bR, a0M, a0R, a1M, a1R;
        bL.x = b.x; bL.y = b.y;
        bR.x = b.z; bR.y = b.w;
        a0M.x = sMean[nc][ak];      a0M.y = sMean[nc][ak + 1];
        a1M.x = sMean[16 + nc][ak]; a1M.y = sMean[16 + nc][ak + 1];
        a0R.x = sRoot[nc][ak];      a0R.y = sRoot[nc][ak + 1];
        a1R.x = sRoot[16 + nc][ak]; a1R.y = sRoot[16 + nc][ak + 1];
        c0 = __builtin_amdgcn_wmma_f32_16x16x4_f32(false, a0M, false, bL,
                                                   (short)0, c0, false, false);
        c1 = __builtin_amdgcn_wmma_f32_16x16x4_f32(false, a1M, false, bL,
                                                   (short)0, c1, false, false);
        c0 = __builtin_amdgcn_wmma_f32_16x16x4_f32(false, a0R, false, bR,
                                                   (short)0, c0, false, false);
        c1 = __builtin_amdgcn_wmma_f32_16x16x4_f32(false, a1R, false, bR,
                                                   (short)0, c1, false, false);
    }

    if (col < ncols) {
        const float badd = bias ? bias[col] : 0.0f;
        #pragma unroll
        for (int v = 0; v < 8; ++v) {
            int r0 = row0 + v + 8 * hi;
            int r1 = r0 + 16;
            if (r0 < nrows) out[(size_t)r0 * ncols + col] = c0[v] + badd;
            if (r1 < nrows) out[(size_t)r1 * ncols + col] = c1[v] + badd;
        }
    }
}

// ---------------------------------------------------------------------------
// per-channel sum & sum-of-squares over rows -> stats[0:128], stats[128:256]
__global__ __launch_bounds__(128)
void bn_stats(const float* __restrict__ h, float* __restrict__ stats,
              int nrows, int rows_per_block) {
    const int ch = threadIdx.x;   // 0..127
    int rbeg = blockIdx.x * rows_per_block;
    int rend = min(rbeg + rows_per_block, nrows);
    float s = 0.f, sq = 0.f;
    for (int r = rbeg; r < rend; ++r) {
        float v = h[(size_t)r * HID_D + ch];
        s += v;
        sq += v * v;
    }
    atomAddF(&stats[ch], s);
    atomAddF(&stats[HID_D + ch], sq);
}

__global__ void bn_apply_relu(float* __restrict__ h, const float* __restrict__ stats,
                              const float* __restrict__ gamma,
                              const float* __restrict__ beta, int nrows) {
    size_t idx = (size_t)blockIdx.x * blockDim.x + threadIdx.x;
    if (idx >= (size_t)nrows * HID_D) return;
    int ch = (int)(idx & (HID_D - 1));
    float invN = 1.0f / (float)nrows;
    float mu  = stats[ch] * invN;
    float var = stats[HID_D + ch] * invN - mu * mu;   // biased variance
    float v = (h[idx] - mu) * rsqrtf(var + BN_EPS) * gamma[ch] + beta[ch];
    h[idx] = v > 0.f ? v : 0.f;
}

// ---------------------------------------------------------------------------
extern "C" void kernel_launch(void* const* d_in, const int* in_sizes, int n_in,
                              void* d_out, int out_size, void* d_ws, size_t ws_size,
                              hipStream_t stream) {
    const float*     x   = (const float*)d_in[0];
    const long long* ei  = (const long long*)d_in[1];   // int64 edge_index [2,E]
    const float*     Wl0 = (const float*)d_in[2];
    const float*     Wr0 = (const float*)d_in[3];
    const float*     Wl1 = (const float*)d_in[4];
    const float*     Wr1 = (const float*)d_in[5];
    const float*     Wl2 = (const float*)d_in[6];
    const float*     Wr2 = (const float*)d_in[7];
    const float*     b2  = (const float*)d_in[8];
    const float*     g0  = (const float*)d_in[9];
    const float*     be0 = (const float*)d_in[10];
    const float*     g1  = (const float*)d_in[11];
    const float*     be1 = (const float*)d_in[12];

    const int N = in_sizes[0] / IN_D;
    const int E = in_sizes[1] / 2;
    const int THID = HID_D / 16;              // 8 column tiles
    const int TOUT = (OUT_D + 15) / 16;       // 3 column tiles

    // workspace carve-up (~155 MB)
    char* ws = (char*)d_ws;
    size_t off = 0;
    auto take = [&](size_t bytes) -> void* {
        void* p = ws + off;
        off = (off + bytes + 255) & ~(size_t)255;
        return p;
    };
    float*  inv   = (float*)take((size_t)N * 4);
    float*  stats = (float*)take(2 * HID_D * 4);
    float4* pk0   = (float4*)take((size_t)THID * KSTEPS * 32 * 16);
    float4* pk1   = (float4*)take((size_t)THID * KSTEPS * 32 * 16);
    float4* pk2   = (float4*)take((size_t)TOUT * KSTEPS * 32 * 16);
    float*  agg   = (float*)take((size_t)N * HID_D * 4);
    float*  hA    = (float*)take((size_t)N * HID_D * 4);
    float*  hB    = (float*)take((size_t)N * HID_D * 4);
    float*  outp  = (float*)d_out;

    const size_t featN = (size_t)N * HID_D;
    const size_t scatN = (size_t)E * (HID_D / 4);
    const int ROWS_PB = 512;
    dim3 gemmBlk(32, THID);                   // 256 threads
    dim3 gemmBlkOut(32, TOUT);                // 96 threads
    int  gemmGrid = (N + 31) / 32;

    // one-time: degree -> 1/max(deg,1), weight fragment packing
    zero_f32<<<(N + 255) / 256, 256, 0, stream>>>(inv, (size_t)N);
    count_deg<<<(E + 255) / 256, 256, 0, stream>>>(ei, inv, E);
    make_inv<<<(N + 255) / 256, 256, 0, stream>>>(inv, N);
    pack_w<<<(THID * KSTEPS * 32 + 255) / 256, 256, 0, stream>>>(Wl0, Wr0, pk0, THID, HID_D);
    pack_w<<<(THID * KSTEPS * 32 + 255) / 256, 256, 0, stream>>>(Wl1, Wr1, pk1, THID, HID_D);
    pack_w<<<(TOUT * KSTEPS * 32 + 255) / 256, 256, 0, stream>>>(Wl2, Wr2, pk2, TOUT, OUT_D);

    // ---- layer 0: x -> hA ----
    zero_f32<<<(featN + 255) / 256, 256, 0, stream>>>(agg, featN);
    scatter_add<<<(scatN + 255) / 256, 256, 0, stream>>>(x, ei, agg, E);
    sage_gemm_wmma<<<gemmGrid, gemmBlk, 0, stream>>>(agg, inv, x, pk0,
                                                     nullptr, hA, N, HID_D);
    zero_f32<<<1, 256, 0, stream>>>(stats, 2 * HID_D);
    bn_stats<<<(N + ROWS_PB - 1) / ROWS_PB, 128, 0, stream>>>(hA, stats, N, ROWS_PB);
    bn_apply_relu<<<(featN + 255) / 256, 256, 0, stream>>>(hA, stats, g0, be0, N);

    // ---- layer 1: hA -> hB ----
    zero_f32<<<(featN + 255) / 256, 256, 0, stream>>>(agg, featN);
    scatter_add<<<(scatN + 255) / 256, 256, 0, stream>>>(hA, ei, agg, E);
    sage_gemm_wmma<<<gemmGrid, gemmBlk, 0, stream>>>(agg, inv, hA, pk1,
                                                     nullptr, hB, N, HID_D);
    zero_f32<<<1, 256, 0, stream>>>(stats, 2 * HID_D);
    bn_stats<<<(N + ROWS_PB - 1) / ROWS_PB, 128, 0, stream>>>(hB, stats, N, ROWS_PB);
    bn_apply_relu<<<(featN + 255) / 256, 256, 0, stream>>>(hB, stats, g1, be1, N);

    // ---- layer 2: hB -> out (bias, no BN/ReLU) ----
    zero_f32<<<(featN + 255) / 256, 256, 0, stream>>>(agg, featN);
    scatter_add<<<(scatN + 255) / 256, 256, 0, stream>>>(hB, ei, agg, E);
    sage_gemm_wmma<<<gemmGrid, gemmBlkOut, 0, stream>>>(agg, inv, hB, pk2,
                                                        b2, outp, N, OUT_D);
}